// AFNO2D_81793357185865
// MI455X (gfx1250) — compile-verified
//
#include <hip/hip_runtime.h>
#include <math.h>

// CDNA5 / gfx1250, wave32. All matrix math via V_WMMA_F32_16X16X4_F32.
typedef float v2f __attribute__((ext_vector_type(2)));
typedef float v8f __attribute__((ext_vector_type(8)));

static __device__ __forceinline__ v8f wmma4(v2f a, v2f b, v8f c) {
  // D = A(16x4) * B(4x16) + C(16x16), fp32
  return __builtin_amdgcn_wmma_f32_16x16x4_f32(false, a, false, b, (short)0, c, false, false);
}

// ---------------------------------------------------------------------------
// K1: forward DFT over w.  Per (b,h): T[64,512] = F(64x256) * x(256x512)
// F rows 0..31 = cos(2pi v w/256)/16, rows 32..63 = -sin(...)/16
// T layout: [b][v][reim][h][c]
// ---------------------------------------------------------------------------
__global__ __launch_bounds__(128) void afno_k1_dft_w(const float* __restrict__ x,
                                                     float* __restrict__ T) {
  __shared__ float tab[256];
  __shared__ float As[64 * 68];
  __shared__ float Bs[64 * 68];
  const int tid = threadIdx.x;
  const int lane = tid & 31, wid = tid >> 5;
  const int l15 = lane & 15, khalf = (lane >> 4) * 2, rbase = (lane >> 4) * 8;
  const int c0 = blockIdx.x * 64;
  const int bh = blockIdx.y;  // b*256 + h

  for (int t = tid; t < 256; t += 128)
    tab[t] = cospif((float)t * (1.0f / 128.0f)) * 0.0625f;

  v8f acc[4];
#pragma unroll
  for (int i = 0; i < 4; ++i)
#pragma unroll
    for (int j = 0; j < 8; ++j) acc[i][j] = 0.0f;

  const float* xbase = x + (size_t)bh * (256 * 512) + c0;

  for (int kc = 0; kc < 4; ++kc) {
    const int k0 = kc * 64;
    __syncthreads();
    for (int j = tid; j < 4096; j += 128) {  // A chunk 64x64
      const int m = j >> 6, kk = j & 63;
      const int w = k0 + kk;
      const int v = m & 31;
      const int off = (m < 32) ? 0 : 64;  // cos : -sin
      As[m * 68 + kk] = tab[(v * w + off) & 255];
    }
    for (int j = tid; j < 1024; j += 128) {  // B chunk 64x64 (f4)
      const int r = j >> 4, c4 = (j & 15) * 4;
      *(float4*)&Bs[r * 68 + c4] = *(const float4*)(xbase + (size_t)(k0 + r) * 512 + c4);
    }
    __syncthreads();
#pragma unroll
    for (int ks = 0; ks < 16; ++ks) {
      const int kb = ks * 4 + khalf;
      v2f bf;
      { const float* p = &Bs[kb * 68 + wid * 16 + l15]; bf.x = p[0]; bf.y = p[68]; }
#pragma unroll
      for (int mt = 0; mt < 4; ++mt) {
        v2f af;
        { const float* p = &As[(mt * 16 + l15) * 68 + kb]; af.x = p[0]; af.y = p[1]; }
        acc[mt] = wmma4(af, bf, acc[mt]);
      }
    }
  }
  const int b = bh >> 8, h = bh & 255;
  const int col = wid * 16 + l15;
#pragma unroll
  for (int mt = 0; mt < 4; ++mt)
#pragma unroll
    for (int r = 0; r < 8; ++r) {
      const int m = mt * 16 + rbase + r;
      const int v = m & 31, reim = m >> 5;
      T[((((size_t)b * 32 + v) * 2 + reim) * 256 + h) * 512 + c0 + col] = acc[mt][r];
    }
}

// ---------------------------------------------------------------------------
// K2: forward DFT over h. Per (b,v): Xh[64,512] = G(64x512) * T(512x512)
// Xre = sum Tre*cos + Tim*sin ; Xim = sum Tim*cos - Tre*sin
// Xh layout: [b][u][v][reim][c]
// ---------------------------------------------------------------------------
__global__ __launch_bounds__(128) void afno_k2_dft_h(const float* __restrict__ T,
                                                     float* __restrict__ Xh) {
  __shared__ float tab[256];
  __shared__ float As[64 * 68];
  __shared__ float Bs[64 * 68];
  const int tid = threadIdx.x;
  const int lane = tid & 31, wid = tid >> 5;
  const int l15 = lane & 15, khalf = (lane >> 4) * 2, rbase = (lane >> 4) * 8;
  const int c0 = blockIdx.x * 64;
  const int bv = blockIdx.y;  // b*32 + v
  const int b = bv >> 5, v = bv & 31;

  for (int t = tid; t < 256; t += 128)
    tab[t] = cospif((float)t * (1.0f / 128.0f)) * 0.0625f;

  v8f acc[4];
#pragma unroll
  for (int i = 0; i < 4; ++i)
#pragma unroll
    for (int j = 0; j < 8; ++j) acc[i][j] = 0.0f;

  const float* Tb = T + (size_t)bv * 262144 + c0;  // rows (reim,h), stride 512

  for (int kc = 0; kc < 8; ++kc) {
    const int k0 = kc * 64;
    __syncthreads();
    for (int j = tid; j < 4096; j += 128) {
      const int m = j >> 6, kk = j & 63;
      const int kg = k0 + kk;
      const int h = kg & 255, hf = kg >> 8;  // hf: 0=Tre rows, 1=Tim rows
      const int um = m & 31;
      const int off = (m < 32) ? (hf ? 192 : 0)   // +sin : cos
                               : (hf ? 0 : 64);   // cos  : -sin
      As[m * 68 + kk] = tab[(um * h + off) & 255];
    }
    for (int j = tid; j < 1024; j += 128) {
      const int r = j >> 4, c4 = (j & 15) * 4;
      *(float4*)&Bs[r * 68 + c4] = *(const float4*)(Tb + (size_t)(k0 + r) * 512 + c4);
    }
    __syncthreads();
#pragma unroll
    for (int ks = 0; ks < 16; ++ks) {
      const int kb = ks * 4 + khalf;
      v2f bf;
      { const float* p = &Bs[kb * 68 + wid * 16 + l15]; bf.x = p[0]; bf.y = p[68]; }
#pragma unroll
      for (int mt = 0; mt < 4; ++mt) {
        v2f af;
        { const float* p = &As[(mt * 16 + l15) * 68 + kb]; af.x = p[0]; af.y = p[1]; }
        acc[mt] = wmma4(af, bf, acc[mt]);
      }
    }
  }
  const int col = wid * 16 + l15;
#pragma unroll
  for (int mt = 0; mt < 4; ++mt)
#pragma unroll
    for (int r = 0; r < 8; ++r) {
      const int m = mt * 16 + rbase + r;
      const int u = m & 31, reim = m >> 5;
      Xh[((((size_t)b * 32 + u) * 32 + v) * 2 + reim) * 512 + c0 + col] = acc[mt][r];
    }
}

// ---------------------------------------------------------------------------
// K3: block-diagonal complex MLP with exact GELU.
// Per block n: O = X(16pos x 128) * B1(128x128); gelu; * B2(128x128)
// B = [[Wr, Wi], [-Wi, Wr]], bias big-vectors [br ; bi].
// ---------------------------------------------------------------------------
__global__ __launch_bounds__(128) void afno_k3_mlp(const float* __restrict__ Xh,
                                                   const float* __restrict__ w1,
                                                   const float* __restrict__ b1,
                                                   const float* __restrict__ w2,
                                                   const float* __restrict__ b2,
                                                   float* __restrict__ S) {
  __shared__ float B1s[128 * 132];
  __shared__ float B2s[128 * 132];
  __shared__ float O1s[4 * 16 * 132];
  const int tid = threadIdx.x;
  const int lane = tid & 31, wid = tid >> 5;
  const int l15 = lane & 15, khalf = (lane >> 4) * 2, rbase = (lane >> 4) * 8;
  const int blk = blockIdx.x;  // 0..7

  const float* W1r = w1 + (size_t)blk * 4096;
  const float* W1i = w1 + (size_t)(8 + blk) * 4096;
  const float* W2r = w2 + (size_t)blk * 4096;
  const float* W2i = w2 + (size_t)(8 + blk) * 4096;
  for (int j = tid; j < 4096; j += 128) {
    const int i = j >> 6, o = j & 63;
    const float a1 = W1r[j], c1 = W1i[j];
    B1s[i * 132 + o] = a1;
    B1s[(64 + i) * 132 + o] = -c1;
    B1s[i * 132 + 64 + o] = c1;
    B1s[(64 + i) * 132 + 64 + o] = a1;
    const float a2 = W2r[j], c2 = W2i[j];
    B2s[i * 132 + o] = a2;
    B2s[(64 + i) * 132 + o] = -c2;
    B2s[i * 132 + 64 + o] = c2;
    B2s[(64 + i) * 132 + 64 + o] = a2;
  }
  __syncthreads();

  const int p0 = blockIdx.y * 64 + wid * 16;  // wave's 16 positions
  const float* Ap = Xh + (size_t)p0 * 1024;

  v8f acc1[8];
#pragma unroll
  for (int i = 0; i < 8; ++i)
#pragma unroll
    for (int j = 0; j < 8; ++j) acc1[i][j] = 0.0f;

  for (int ks = 0; ks < 32; ++ks) {
    const int kb = ks * 4 + khalf;
    const int g = (kb < 64) ? (blk * 64 + kb) : (448 + blk * 64 + kb);  // xr / xi
    v2f af;
    { const float* p = Ap + (size_t)l15 * 1024 + g; af.x = p[0]; af.y = p[1]; }
#pragma unroll
    for (int nt = 0; nt < 8; ++nt) {
      v2f bf;
      { const float* p = &B1s[kb * 132 + nt * 16 + l15]; bf.x = p[0]; bf.y = p[132]; }
      acc1[nt] = wmma4(af, bf, acc1[nt]);
    }
  }
  float* O1w = &O1s[wid * 16 * 132];
#pragma unroll
  for (int nt = 0; nt < 8; ++nt) {
    const int col = nt * 16 + l15;
    const float bias = (col < 64) ? b1[blk * 64 + col] : b1[448 + blk * 64 + col];
#pragma unroll
    for (int r = 0; r < 8; ++r) {
      const float xv = acc1[nt][r] + bias;
      const float gv = 0.5f * xv * (1.0f + erff(xv * 0.70710678118654752f));
      O1w[(rbase + r) * 132 + col] = gv;
    }
  }
  __syncthreads();

  v8f acc2[8];
#pragma unroll
  for (int i = 0; i < 8; ++i)
#pragma unroll
    for (int j = 0; j < 8; ++j) acc2[i][j] = 0.0f;

  for (int ks = 0; ks < 32; ++ks) {
    const int kb = ks * 4 + khalf;
    v2f af;
    { const float* p = &O1w[l15 * 132 + kb]; af.x = p[0]; af.y = p[1]; }
#pragma unroll
    for (int nt = 0; nt < 8; ++nt) {
      v2f bf;
      { const float* p = &B2s[kb * 132 + nt * 16 + l15]; bf.x = p[0]; bf.y = p[132]; }
      acc2[nt] = wmma4(af, bf, acc2[nt]);
    }
  }
#pragma unroll
  for (int nt = 0; nt < 8; ++nt) {
    const int col = nt * 16 + l15;
    const int g = (col < 64) ? (blk * 64 + col) : (448 + blk * 64 + col);
    const float bias = (col < 64) ? b2[blk * 64 + col] : b2[448 + blk * 64 + col];
#pragma unroll
    for (int r = 0; r < 8; ++r) {
      const int p = p0 + rbase + r;
      S[(size_t)p * 1024 + g] = acc2[nt][r] + bias;
    }
  }
}

// ---------------------------------------------------------------------------
// K4: inverse DFT over h (32 modes -> 256 rows), e^{+i}:
// Zre = sum Sre*cos - Sim*sin ; Zim = sum Sre*sin + Sim*cos
// Z layout: [b][h][reim][v][c]
// ---------------------------------------------------------------------------
__global__ __launch_bounds__(128) void afno_k4_idft_h(const float* __restrict__ S,
                                                      float* __restrict__ Z) {
  __shared__ float tab[256];
  __shared__ float As[64 * 68];
  __shared__ float Bs[64 * 68];
  const int tid = threadIdx.x;
  const int lane = tid & 31, wid = tid >> 5;
  const int l15 = lane & 15, khalf = (lane >> 4) * 2, rbase = (lane >> 4) * 8;
  const int c0 = blockIdx.x * 64;
  const int h0 = blockIdx.y * 32;
  const int bv = blockIdx.z;
  const int b = bv >> 5, v = bv & 31;

  for (int t = tid; t < 256; t += 128)
    tab[t] = cospif((float)t * (1.0f / 128.0f)) * 0.0625f;
  __syncthreads();

  for (int j = tid; j < 4096; j += 128) {
    const int m = j >> 6, kk = j & 63;
    const int h = h0 + (m & 31);
    const int u = kk & 31;
    const int off = (m < 32) ? ((kk < 32) ? 0 : 64)     // cos : -sin
                             : ((kk < 32) ? 192 : 0);   // sin : cos
    As[m * 68 + kk] = tab[(u * h + off) & 255];
  }
  for (int j = tid; j < 1024; j += 128) {
    const int r = j >> 4, c4 = (j & 15) * 4;
    const int u = r & 31, reim = r >> 5;
    *(float4*)&Bs[r * 68 + c4] =
        *(const float4*)(S + ((((size_t)b * 32 + u) * 32 + v) * 2 + reim) * 512 + c0 + c4);
  }
  __syncthreads();

  v8f acc[4];
#pragma unroll
  for (int i = 0; i < 4; ++i)
#pragma unroll
    for (int j = 0; j < 8; ++j) acc[i][j] = 0.0f;

#pragma unroll
  for (int ks = 0; ks < 16; ++ks) {
    const int kb = ks * 4 + khalf;
    v2f bf;
    { const float* p = &Bs[kb * 68 + wid * 16 + l15]; bf.x = p[0]; bf.y = p[68]; }
#pragma unroll
    for (int mt = 0; mt < 4; ++mt) {
      v2f af;
      { const float* p = &As[(mt * 16 + l15) * 68 + kb]; af.x = p[0]; af.y = p[1]; }
      acc[mt] = wmma4(af, bf, acc[mt]);
    }
  }
  const int col = wid * 16 + l15;
#pragma unroll
  for (int mt = 0; mt < 4; ++mt)
#pragma unroll
    for (int r = 0; r < 8; ++r) {
      const int m = mt * 16 + rbase + r;
      const int reim = m >> 5, h = h0 + (m & 31);
      Z[((((size_t)b * 256 + h) * 2 + reim) * 32 + v) * 512 + c0 + col] = acc[mt][r];
    }
}

// ---------------------------------------------------------------------------
// K5: Hermitian inverse DFT over w + residual add.
// y[w] = x[w] + sum_v alpha_v (Zre*cos(2pi v w/256) - Zim*sin)/16
// ---------------------------------------------------------------------------
__global__ __launch_bounds__(256) void afno_k5_idft_w(const float* __restrict__ Z,
                                                      const float* __restrict__ x,
                                                      float* __restrict__ y) {
  __shared__ float tab[256];
  __shared__ float As[256 * 68];
  __shared__ float Bs[64 * 68];
  const int tid = threadIdx.x;
  const int lane = tid & 31, wid = tid >> 5;  // 8 waves
  const int l15 = lane & 15, khalf = (lane >> 4) * 2, rbase = (lane >> 4) * 8;
  const int c0 = blockIdx.x * 64;
  const int bh = blockIdx.y;

  for (int t = tid; t < 256; t += 256)
    tab[t] = cospif((float)t * (1.0f / 128.0f)) * 0.0625f;
  __syncthreads();

  for (int j = tid; j < 16384; j += 256) {
    const int m = j >> 6, kk = j & 63;  // m = w
    const int v = kk & 31;
    const int off = (kk < 32) ? 0 : 64;  // cos : -sin
    const float alpha = (v == 0) ? 1.0f : 2.0f;
    As[m * 68 + kk] = alpha * tab[(v * m + off) & 255];
  }
  const float* Zb = Z + (size_t)bh * 32768 + c0;
  for (int j = tid; j < 1024; j += 256) {
    const int r = j >> 4, c4 = (j & 15) * 4;
    *(float4*)&Bs[r * 68 + c4] = *(const float4*)(Zb + (size_t)r * 512 + c4);
  }
  __syncthreads();

  v8f acc[2][4];
#pragma unroll
  for (int a = 0; a < 2; ++a)
#pragma unroll
    for (int i = 0; i < 4; ++i)
#pragma unroll
      for (int j = 0; j < 8; ++j) acc[a][i][j] = 0.0f;

#pragma unroll
  for (int ks = 0; ks < 16; ++ks) {
    const int kb = ks * 4 + khalf;
    v2f bf[4];
#pragma unroll
    for (int nt = 0; nt < 4; ++nt) {
      const float* p = &Bs[kb * 68 + nt * 16 + l15];
      bf[nt].x = p[0];
      bf[nt].y = p[68];
    }
#pragma unroll
    for (int jm = 0; jm < 2; ++jm) {
      const int mt = wid * 2 + jm;
      v2f af;
      { const float* p = &As[(mt * 16 + l15) * 68 + kb]; af.x = p[0]; af.y = p[1]; }
#pragma unroll
      for (int nt = 0; nt < 4; ++nt) acc[jm][nt] = wmma4(af, bf[nt], acc[jm][nt]);
    }
  }
  const size_t obase = (size_t)bh * (256 * 512) + c0;
#pragma unroll
  for (int jm = 0; jm < 2; ++jm)
#pragma unroll
    for (int nt = 0; nt < 4; ++nt) {
      const int col = nt * 16 + l15;
#pragma unroll
      for (int r = 0; r < 8; ++r) {
        const int m = (wid * 2 + jm) * 16 + rbase + r;
        const size_t idx = obase + (size_t)m * 512 + col;
        y[idx] = acc[jm][nt][r] + x[idx];
      }
    }
}

// ---------------------------------------------------------------------------
extern "C" void kernel_launch(void* const* d_in, const int* in_sizes, int n_in,
                              void* d_out, int out_size, void* d_ws, size_t ws_size,
                              hipStream_t stream) {
  (void)in_sizes; (void)n_in; (void)out_size; (void)ws_size;
  const float* x  = (const float*)d_in[0];
  const float* w1 = (const float*)d_in[1];
  const float* b1 = (const float*)d_in[2];
  const float* w2 = (const float*)d_in[3];
  const float* b2 = (const float*)d_in[4];
  float* out = (float*)d_out;
  float* ws = (float*)d_ws;

  float* T  = ws;               // 16,777,216 floats: T (K1/K2), reused as Z (K4/K5)
  float* Xh = ws + 16777216;    //  2,097,152 floats: spectrum corner
  float* Sh = ws + 18874368;    //  2,097,152 floats: MLP output

  afno_k1_dft_w <<<dim3(8, 512),    128, 0, stream>>>(x, T);
  afno_k2_dft_h <<<dim3(8, 64),     128, 0, stream>>>(T, Xh);
  afno_k3_mlp   <<<dim3(8, 32),     128, 0, stream>>>(Xh, w1, b1, w2, b2, Sh);
  afno_k4_idft_h<<<dim3(8, 8, 64),  128, 0, stream>>>(Sh, T);
  afno_k5_idft_w<<<dim3(8, 512),    256, 0, stream>>>(T, x, out);
}